// GAT_16080357556478
// MI455X (gfx1250) — compile-verified
//
#include <hip/hip_runtime.h>
#include <math.h>

#define N_NODES 50000
#define N_EDGES 800000
#define E_TOT   (N_EDGES + N_NODES)   // edges + self loops
#define D 128
#define NEG_SLOPE 0.2f

typedef float v2f __attribute__((ext_vector_type(2)));
typedef float v8f __attribute__((ext_vector_type(8)));

// ---------------------------------------------------------------------------
// C[nrows x 128] = A[nrows x 128] * W[128 x 128]  (row major), fp32 WMMA.
// One wave -> one 16x16 output tile; block = 256 threads = 8 waves covering
// one 16-row strip across all 128 output columns.
// A-frag (16x4 f32): lane L<16 holds A[M=L][k0+0..1], lane L>=16 A[M=L-16][k0+2..3]
// B-frag (4x16 f32): lane L<16 holds W[k0+0..1][N=L], lane L>=16 W[k0+2..3][N=L-16]
// D/C (16x16 f32):   vgpr r, lanes0-15 -> M=r, lanes16-31 -> M=r+8, N=lane&15
// ---------------------------------------------------------------------------
__global__ void gemm_wmma_f32(const float* __restrict__ A,
                              const float* __restrict__ W,
                              float* __restrict__ C, int nrows) {
  const int wave  = threadIdx.x >> 5;
  const int lane  = threadIdx.x & 31;
  const int m0    = blockIdx.x * 16;
  const int n0    = wave * 16;
  const int sub   = lane & 15;   // row of A-frag / col of B-frag
  const int khalf = lane >> 4;   // 0 or 1
  if (m0 >= nrows) return;

  v8f acc = {};
  const float* arow = A + (size_t)(m0 + sub) * D;
  const float* wcol = W + n0 + sub;

#pragma unroll
  for (int k0 = 0; k0 < D; k0 += 4) {
    const int ka = k0 + 2 * khalf;
    v2f a, b;
    a.x = arow[ka];
    a.y = arow[ka + 1];
    b.x = wcol[(size_t)ka * D];
    b.y = wcol[(size_t)(ka + 1) * D];
    acc = __builtin_amdgcn_wmma_f32_16x16x4_f32(
        /*neg_a=*/false, a, /*neg_b=*/false, b,
        /*c_mod=*/(short)0, acc, /*reuse_a=*/false, /*reuse_b=*/false);
  }

  float* cptr = C + n0 + sub;
#pragma unroll
  for (int r = 0; r < 8; ++r) {
    const int m = m0 + r + 8 * khalf;
    cptr[(size_t)m * D] = acc[r];
  }
}

// ---------------------------------------------------------------------------
// edge weights: ew = attr==0 ? 1e4 : 1/attr ; self loops -> 0
// ---------------------------------------------------------------------------
__global__ void ew_kernel(const float* __restrict__ attr, float* __restrict__ ew) {
  const int e = blockIdx.x * blockDim.x + threadIdx.x;
  if (e >= E_TOT) return;
  if (e < N_EDGES) {
    const float a = attr[e];
    ew[e] = (a == 0.0f) ? 10000.0f : (1.0f / a);
  } else {
    ew[e] = 0.0f;
  }
}

// per-layer init: h_next = bias, mKey = encode(-inf), denom = 0
__global__ void init_kernel(float* __restrict__ hnext, const float* __restrict__ bias,
                            unsigned* __restrict__ mKey, float* __restrict__ denom) {
  const int i = blockIdx.x;
  hnext[(size_t)i * D + threadIdx.x] = bias[threadIdx.x];
  if (threadIdx.x == 0) {
    mKey[i]  = 0x007FFFFFu;  // ~bits(-inf): order-preserving key for -inf
    denom[i] = 0.0f;
  }
}

__device__ __forceinline__ void edge_nodes(const int* __restrict__ ei, int e,
                                           int& s, int& d) {
  if (e < N_EDGES) { s = ei[e]; d = ei[N_EDGES + e]; }
  else             { s = e - N_EDGES; d = s; }
}

__device__ __forceinline__ unsigned f32_key(float f) {
  const unsigned b = __float_as_uint(f);
  return (b & 0x80000000u) ? ~b : (b | 0x80000000u);
}
__device__ __forceinline__ float key_f32(unsigned k) {
  return __uint_as_float((k & 0x80000000u) ? (k & 0x7FFFFFFFu) : ~k);
}

// ---------------------------------------------------------------------------
// pass 1: logits[e] = att . leakyrelu(xl[src] + xr[dst] + ew*We)
//         + atomic segment-max via ordered-uint keys. One wave per edge.
// ---------------------------------------------------------------------------
__global__ void edge_logits_kernel(const int* __restrict__ ei, const float* __restrict__ ew,
                                   const float* __restrict__ xl, const float* __restrict__ xr,
                                   const float* __restrict__ We, const float* __restrict__ att,
                                   float* __restrict__ logits, unsigned* __restrict__ mKey) {
  const int e = blockIdx.x * (blockDim.x >> 5) + (threadIdx.x >> 5);
  const int lane = threadIdx.x & 31;
  if (e >= E_TOT) return;
  int s, d; edge_nodes(ei, e, s, d);
  const float w = ew[e];
  const int c = lane * 4;

  const float4 vl = *(const float4*)(xl + (size_t)s * D + c);
  const float4 vr = *(const float4*)(xr + (size_t)d * D + c);
  const float4 ve = *(const float4*)(We + c);
  const float4 va = *(const float4*)(att + c);

  float p = 0.0f, t;
  t = vl.x + vr.x + w * ve.x; t = (t > 0.0f) ? t : NEG_SLOPE * t; p += t * va.x;
  t = vl.y + vr.y + w * ve.y; t = (t > 0.0f) ? t : NEG_SLOPE * t; p += t * va.y;
  t = vl.z + vr.z + w * ve.z; t = (t > 0.0f) ? t : NEG_SLOPE * t; p += t * va.z;
  t = vl.w + vr.w + w * ve.w; t = (t > 0.0f) ? t : NEG_SLOPE * t; p += t * va.w;

#pragma unroll
  for (int off = 16; off > 0; off >>= 1) p += __shfl_xor(p, off, 32);

  if (lane == 0) {
    logits[e] = p;
    atomicMax(mKey + d, f32_key(p));
  }
}

// pass 2: e = exp(logit - m[dst]); denom[dst] += e.  Thread per edge.
__global__ void edge_exp_kernel(const int* __restrict__ ei, float* __restrict__ logits,
                                const unsigned* __restrict__ mKey, float* __restrict__ denom) {
  const int e = blockIdx.x * blockDim.x + threadIdx.x;
  if (e >= E_TOT) return;
  int s, d; edge_nodes(ei, e, s, d);
  const float m  = key_f32(mKey[d]);
  const float ex = expf(logits[e] - m);
  logits[e] = ex;
  atomicAdd(denom + d, ex);
}

// pass 3: h_next[dst] += (e/denom[dst]) * xl[src].  One wave per edge.
__global__ void edge_aggr_kernel(const int* __restrict__ ei, const float* __restrict__ logits,
                                 const float* __restrict__ denom, const float* __restrict__ xl,
                                 float* __restrict__ hnext) {
  const int e = blockIdx.x * (blockDim.x >> 5) + (threadIdx.x >> 5);
  const int lane = threadIdx.x & 31;
  if (e >= E_TOT) return;
  int s, d; edge_nodes(ei, e, s, d);
  const float alpha = logits[e] / denom[d];
  const int c = lane * 4;
  const float4 v = *(const float4*)(xl + (size_t)s * D + c);
  float* o = hnext + (size_t)d * D + c;
  atomicAdd(o + 0, alpha * v.x);
  atomicAdd(o + 1, alpha * v.y);
  atomicAdd(o + 2, alpha * v.z);
  atomicAdd(o + 3, alpha * v.w);
}

__global__ void relu_kernel(float* __restrict__ h, int n) {
  const int i = blockIdx.x * blockDim.x + threadIdx.x;
  if (i < n) h[i] = fmaxf(h[i], 0.0f);
}

// out[i] = h[i] . Wlin + blin.  One wave per node.
__global__ void final_lin_kernel(const float* __restrict__ h, const float* __restrict__ Wlin,
                                 const float* __restrict__ blin, float* __restrict__ out) {
  const int i = blockIdx.x * (blockDim.x >> 5) + (threadIdx.x >> 5);
  const int lane = threadIdx.x & 31;
  if (i >= N_NODES) return;
  const int c = lane * 4;
  const float4 v = *(const float4*)(h + (size_t)i * D + c);
  const float4 w = *(const float4*)(Wlin + c);
  float p = v.x * w.x + v.y * w.y + v.z * w.z + v.w * w.w;
#pragma unroll
  for (int off = 16; off > 0; off >>= 1) p += __shfl_xor(p, off, 32);
  if (lane == 0) out[i] = p + blin[0];
}

extern "C" void kernel_launch(void* const* d_in, const int* in_sizes, int n_in,
                              void* d_out, int out_size, void* d_ws, size_t ws_size,
                              hipStream_t stream) {
  const float* x    = (const float*)d_in[0];
  const int*   ei   = (const int*)  d_in[1];
  const float* attr = (const float*)d_in[2];
  const float* Wl[3]  = {(const float*)d_in[3],  (const float*)d_in[8],  (const float*)d_in[13]};
  const float* Wr[3]  = {(const float*)d_in[4],  (const float*)d_in[9],  (const float*)d_in[14]};
  const float* We[3]  = {(const float*)d_in[5],  (const float*)d_in[10], (const float*)d_in[15]};
  const float* att[3] = {(const float*)d_in[6],  (const float*)d_in[11], (const float*)d_in[16]};
  const float* bia[3] = {(const float*)d_in[7],  (const float*)d_in[12], (const float*)d_in[17]};
  const float* Wlin = (const float*)d_in[18];
  const float* blin = (const float*)d_in[19];

  // workspace carve-up (all offsets 16B aligned)
  float* ws = (float*)d_ws;
  size_t off = 0;
  float*    ew     = ws + off; off += E_TOT;
  float*    xl     = ws + off; off += (size_t)N_NODES * D;
  float*    xr     = ws + off; off += (size_t)N_NODES * D;
  float*    logits = ws + off; off += E_TOT;
  unsigned* mKey   = (unsigned*)(ws + off); off += N_NODES;
  float*    denom  = ws + off; off += N_NODES;
  float*    hA     = ws + off; off += (size_t)N_NODES * D;
  float*    hB     = ws + off; off += (size_t)N_NODES * D;

  ew_kernel<<<(E_TOT + 255) / 256, 256, 0, stream>>>(attr, ew);

  const int gemm_blocks = N_NODES / 16;           // 3125 (N divisible by 16)
  const int edge_wave_blocks = (E_TOT + 7) / 8;   // 8 waves per 256-thr block

  const float* layer_in = x;
  float* layer_out = hA;
  for (int l = 0; l < 3; ++l) {
    gemm_wmma_f32<<<gemm_blocks, 256, 0, stream>>>(layer_in, Wl[l], xl, N_NODES);
    gemm_wmma_f32<<<gemm_blocks, 256, 0, stream>>>(layer_in, Wr[l], xr, N_NODES);
    init_kernel<<<N_NODES, 128, 0, stream>>>(layer_out, bia[l], mKey, denom);
    edge_logits_kernel<<<edge_wave_blocks, 256, 0, stream>>>(ei, ew, xl, xr, We[l], att[l],
                                                             logits, mKey);
    edge_exp_kernel<<<(E_TOT + 255) / 256, 256, 0, stream>>>(ei, logits, mKey, denom);
    edge_aggr_kernel<<<edge_wave_blocks, 256, 0, stream>>>(ei, logits, denom, xl, layer_out);
    if (l < 2) {
      relu_kernel<<<((size_t)N_NODES * D + 255) / 256, 256, 0, stream>>>(layer_out,
                                                                         N_NODES * D);
    }
    layer_in  = layer_out;
    layer_out = (l == 0) ? hB : hA;   // l0: x->hA, l1: hA->hB, l2: hB->hA
  }

  final_lin_kernel<<<(N_NODES + 7) / 8, 256, 0, stream>>>(hA, Wlin, blin, (float*)d_out);
}